// FeedForwardAttention_85615878078675
// MI455X (gfx1250) — compile-verified
//
#include <hip/hip_runtime.h>
#include <hip/hip_bf16.h>
#include <cstdint>

typedef float v2f __attribute__((ext_vector_type(2)));
typedef float v8f __attribute__((ext_vector_type(8)));

#if __has_builtin(__builtin_amdgcn_wmma_f32_16x16x4_f32)
#define HAS_WMMA_F32X4 1
#else
#define HAS_WMMA_F32X4 0
#endif

// Problem dims (fixed by the reference)
#define BB 8
#define QQ 128
#define CX 256
#define CC 1024
#define QS 1024
#define HH 1024

// ---------------------------------------------------------------------------
// K1/K2: D[M,N] = A[M,K] * W[N,K]^T (+ bias[n]).  f32 WMMA 16x16x4.
// Block = 256 threads = 8 waves in a 2(M) x 4(N) grid -> 32x64 output tile.
// A frag: lane(half,lr) holds A[m0+lr][k+2*half .. +1]     (contiguous pair)
// B frag: B[k][n] = W[n][k] -> lane holds W[n0+lr][k+2*half .. +1] (contiguous)
// C/D:    VGPR j, lane(half,lr) = D[m0 + j + 8*half][n0+lr]
// ---------------------------------------------------------------------------
__global__ __launch_bounds__(256)
void gemm_awT(const float* __restrict__ A, const float* __restrict__ W,
              const float* __restrict__ bias, float* __restrict__ D,
              int M, int N, int K, int hasBias) {
  const int lane = threadIdx.x & 31;
  const int w    = threadIdx.x >> 5;
  const int wm   = w >> 2, wn = w & 3;
  const int half = lane >> 4, lr = lane & 15;
  const int m0   = blockIdx.x * 32 + wm * 16;
  const int n0   = blockIdx.y * 64 + wn * 16;

  v8f acc = {};
  const float* arow = A + (size_t)(m0 + lr) * K;
  const float* wrow = W + (size_t)(n0 + lr) * K;

  for (int k = 0; k < K; k += 4) {
#if HAS_WMMA_F32X4
    const int kk = k + 2 * half;
    v2f a  = *(const v2f*)(arow + kk);
    v2f bf = *(const v2f*)(wrow + kk);
    acc = __builtin_amdgcn_wmma_f32_16x16x4_f32(false, a, false, bf,
                                                (short)0, acc, false, false);
#else
    #pragma unroll
    for (int j = 0; j < 8; ++j) {
      const int row = m0 + j + 8 * half;
      float s = acc[j];
      #pragma unroll
      for (int t = 0; t < 4; ++t)
        s = fmaf(A[(size_t)row * K + k + t], W[(size_t)(n0 + lr) * K + k + t], s);
      acc[j] = s;
    }
#endif
  }
  const float bv = hasBias ? bias[n0 + lr] : 0.0f;
  #pragma unroll
  for (int j = 0; j < 8; ++j) {
    const int row = m0 + j + 8 * half;
    D[(size_t)row * N + n0 + lr] = acc[j] + bv;
  }
}

// ---------------------------------------------------------------------------
// K3: scores + softmax.  One block per (b, 16-query tile).
// qh tile + wk resident in LDS; ch rows streamed via async global->LDS
// (double buffered, one b128 per thread == one 4KB row per block).
// ---------------------------------------------------------------------------
__global__ __launch_bounds__(256)
void scores_softmax(const float* __restrict__ ch, const float* __restrict__ qh,
                    const float* __restrict__ wk, const unsigned char* __restrict__ mask,
                    float* __restrict__ attnOut /*[Q,B,Cx]*/,
                    float* __restrict__ attnB   /*[B,Q,Cx]*/) {
  __shared__ __align__(16) float qsh[16 * HH];     // 64 KB
  __shared__ __align__(16) float chbuf[2][HH];     //  8 KB
  __shared__ __align__(16) float wksh[HH];         //  4 KB
  __shared__ __align__(16) float ssc[16 * CX];     // 16 KB

  const int b  = blockIdx.x;
  const int q0 = blockIdx.y * 16;
  const int t  = threadIdx.x;
  const int w  = t >> 5, lane = t & 31;

  // Stage qh tile and wk into LDS
  for (int i = t; i < 16 * HH; i += 256) {
    const int qq = i >> 10, hh = i & (HH - 1);
    qsh[i] = qh[((size_t)(q0 + qq) * BB + b) * HH + hh];
  }
  for (int i = t; i < HH; i += 256) wksh[i] = wk[i];

  const float* chb = ch + (size_t)b * CX * HH;

  // Prefetch ch row c=0 asynchronously (each thread: 16 bytes -> full 4KB row)
  {
    const float* g = chb + (size_t)t * 4;
    unsigned l = (unsigned)(uintptr_t)&chbuf[0][t * 4];
    asm volatile("global_load_async_to_lds_b128 %0, %1, off" :: "v"(l), "v"(g) : "memory");
  }
  __syncthreads();  // qsh/wksh visible

  for (int c = 0; c < CX; ++c) {
    if (c + 1 < CX) {
      const float* g = chb + (size_t)(c + 1) * HH + (size_t)t * 4;
      unsigned l = (unsigned)(uintptr_t)&chbuf[(c + 1) & 1][t * 4];
      asm volatile("global_load_async_to_lds_b128 %0, %1, off" :: "v"(l), "v"(g) : "memory");
      asm volatile("s_wait_asynccnt 1" ::: "memory");  // row c done, c+1 in flight
    } else {
      asm volatile("s_wait_asynccnt 0" ::: "memory");
    }
    __syncthreads();  // buffer c visible to all waves

    const float* cb = chbuf[c & 1];
    #pragma unroll
    for (int qq = 0; qq < 2; ++qq) {
      const int q = w * 2 + qq;
      float acc = 0.0f;
      #pragma unroll 4
      for (int j = 0; j < 32; ++j) {
        const int h = j * 32 + lane;
        const float x = qsh[q * HH + h] + cb[h];
        float th;
#if __has_builtin(__builtin_amdgcn_tanhf)
        th = __builtin_amdgcn_tanhf(x);
#else
        th = tanhf(x);
#endif
        acc = fmaf(wksh[h], th, acc);
      }
      #pragma unroll
      for (int off = 16; off > 0; off >>= 1) acc += __shfl_xor(acc, off, 32);
      if (lane == 0) ssc[q * CX + c] = acc;
    }
    __syncthreads();  // protect chbuf[(c+1)&1] before next-iteration stage
  }

  // Fused softmax over c (256 values per (q) row); wave w owns q = 2w, 2w+1
  #pragma unroll
  for (int qq = 0; qq < 2; ++qq) {
    const int q = w * 2 + qq;
    float v[8];
    float m = -__builtin_inff();
    #pragma unroll
    for (int i = 0; i < 8; ++i) {
      const int c = i * 32 + lane;
      const float s = mask[b * CX + c] ? -__builtin_inff() : ssc[q * CX + c];
      v[i] = s;
      m = fmaxf(m, s);
    }
    #pragma unroll
    for (int off = 16; off > 0; off >>= 1) m = fmaxf(m, __shfl_xor(m, off, 32));
    float sum = 0.0f;
    #pragma unroll
    for (int i = 0; i < 8; ++i) { v[i] = __expf(v[i] - m); sum += v[i]; }
    #pragma unroll
    for (int off = 16; off > 0; off >>= 1) sum += __shfl_xor(sum, off, 32);
    const float inv = 1.0f / sum;
    #pragma unroll
    for (int i = 0; i < 8; ++i) {
      const int c = i * 32 + lane;
      const float a = v[i] * inv;
      attnOut[((size_t)(q0 + q) * BB + b) * CX + c] = a;  // output [Q,B,Cx]
      attnB[((size_t)b * QQ + (q0 + q)) * CX + c] = a;    // [B,Q,Cx] for K4
    }
  }
}

// ---------------------------------------------------------------------------
// K4: composition[q,b,:] = attn[b,q,:] * context[b,:,:]   (per-b GEMM, WMMA)
// M=Q=128, K=Cx=256, N=C=1024.  B[k][n] = context_b[k*C + n].
// ---------------------------------------------------------------------------
__global__ __launch_bounds__(256)
void composition_gemm(const float* __restrict__ attnB, const float* __restrict__ ctx,
                      float* __restrict__ outC /*[Q,B,C]*/) {
  const int lane = threadIdx.x & 31;
  const int w    = threadIdx.x >> 5;
  const int wm   = w >> 2, wn = w & 3;
  const int half = lane >> 4, lr = lane & 15;
  const int b    = blockIdx.z;
  const int m0   = blockIdx.x * 32 + wm * 16;   // q tile
  const int n0   = blockIdx.y * 64 + wn * 16;   // d tile

  const float* Ab = attnB + (size_t)b * QQ * CX;
  const float* Cb = ctx + (size_t)b * CX * CC;

  v8f acc = {};
  for (int k = 0; k < CX; k += 4) {
#if HAS_WMMA_F32X4
    const int kk = k + 2 * half;
    v2f a = *(const v2f*)(Ab + (size_t)(m0 + lr) * CX + kk);
    v2f bf;
    bf.x = Cb[(size_t)kk * CC + n0 + lr];
    bf.y = Cb[(size_t)(kk + 1) * CC + n0 + lr];
    acc = __builtin_amdgcn_wmma_f32_16x16x4_f32(false, a, false, bf,
                                                (short)0, acc, false, false);
#else
    #pragma unroll
    for (int j = 0; j < 8; ++j) {
      const int row = m0 + j + 8 * half;
      float s = acc[j];
      #pragma unroll
      for (int tkt = 0; tkt < 4; ++tkt)
        s = fmaf(Ab[(size_t)row * CX + k + tkt], Cb[(size_t)(k + tkt) * CC + n0 + lr], s);
      acc[j] = s;
    }
#endif
  }
  #pragma unroll
  for (int j = 0; j < 8; ++j) {
    const int q = m0 + j + 8 * half;
    outC[((size_t)q * BB + b) * CC + n0 + lr] = acc[j];
  }
}

// ---------------------------------------------------------------------------
extern "C" void kernel_launch(void* const* d_in, const int* in_sizes, int n_in,
                              void* d_out, int out_size, void* d_ws, size_t ws_size,
                              hipStream_t stream) {
  const float* context        = (const float*)d_in[0];          // [B,Cx,C]
  const float* query          = (const float*)d_in[1];          // [Q,B,Qs]
  const unsigned char* mask   = (const unsigned char*)d_in[2];  // [B,Cx] bool
  const float* Wc             = (const float*)d_in[3];          // [H,C]
  const float* Wq             = (const float*)d_in[4];          // [H,Qs]
  const float* bq             = (const float*)d_in[5];          // [H]
  const float* wk             = (const float*)d_in[6];          // [H]

  float* out     = (float*)d_out;
  float* attnOut = out;                          // [Q,B,Cx]
  float* comp    = out + (size_t)QQ * BB * CX;   // [Q,B,C]

  float* ch    = (float*)d_ws;                       // [B*Cx, H]  8 MB
  float* qh    = ch + (size_t)BB * CX * HH;          // [Q*B, H]   4 MB
  float* attnB = qh + (size_t)QQ * BB * HH;          // [B,Q,Cx]   1 MB

  dim3 blk(256);
  // ch = context * Wc^T
  gemm_awT<<<dim3((BB * CX) / 32, HH / 64), blk, 0, stream>>>(
      context, Wc, nullptr, ch, BB * CX, HH, CC, 0);
  // qh = query * Wq^T + bq
  gemm_awT<<<dim3((QQ * BB) / 32, HH / 64), blk, 0, stream>>>(
      query, Wq, bq, qh, QQ * BB, HH, QS, 1);
  // scores + softmax -> attention
  scores_softmax<<<dim3(BB, QQ / 16), blk, 0, stream>>>(
      ch, qh, wk, mask, attnOut, attnB);
  // composition = attention * context
  composition_gemm<<<dim3(QQ / 32, CC / 64, BB), blk, 0, stream>>>(
      attnB, context, comp);
}